// GraphConvolution_9577777070215
// MI455X (gfx1250) — compile-verified
//
#include <hip/hip_runtime.h>

#define D_FEAT 96

typedef __attribute__((ext_vector_type(2))) float v2f;
typedef __attribute__((ext_vector_type(8))) float v8f;

// ---------------------------------------------------------------------------
// Kernel 1: zero the accumulator (d_out is reused as hidden_hat scratch).
// ---------------------------------------------------------------------------
__global__ void gcn_zero_f4(float4* __restrict__ p, int n4) {
    int i = blockIdx.x * blockDim.x + threadIdx.x;
    int stride = gridDim.x * blockDim.x;
    for (; i < n4; i += stride) {
        p[i] = make_float4(0.f, 0.f, 0.f, 0.f);
    }
}

// ---------------------------------------------------------------------------
// Kernel 2: one wave32 per edge. Lanes stride the 96 features:
// coalesced gather of x[col], scale by edge_w, non-returning f32 atomic add
// into hidden_hat[row]. hidden_hat (19.2 MB) is L2-resident on MI455X.
// ---------------------------------------------------------------------------
__global__ void __launch_bounds__(256)
gcn_edge_scatter(const float* __restrict__ x,
                 const float* __restrict__ ew,
                 const int*   __restrict__ row,
                 const int*   __restrict__ col,
                 float*       __restrict__ hh,
                 int E) {
    int wid  = (blockIdx.x * blockDim.x + threadIdx.x) >> 5;
    int lane = threadIdx.x & 31;
    if (wid >= E) return;

    int   r = row[wid];
    int   c = col[wid];
    float w = ew[wid];

    const float* xs = x  + (size_t)c * D_FEAT;
    float*       hs = hh + (size_t)r * D_FEAT;

#pragma unroll
    for (int f = lane; f < D_FEAT; f += 32) {
        __hip_atomic_fetch_add(&hs[f], w * xs[f],
                               __ATOMIC_RELAXED, __HIP_MEMORY_SCOPE_AGENT);
    }
}

// ---------------------------------------------------------------------------
// Kernel 3: in-place GEMM + bias + ReLU using V_WMMA_F32_16X16X4_F32.
// One wave32 owns a 16-row stripe: preloads A = hh[16 x 96] into registers
// (24 k-chunks of 4, per the CDNA5 f32 A layout), then produces six 16x16
// output tiles (96 cols), 24 WMMAs each. Since the full A stripe is held in
// VGPRs before any store, the update is safely in place on d_out.
//
// A layout (16x4 f32): lane l<16 holds A[m=l][K=4kk+0], [4kk+1];
//                      lane l>=16 holds A[m=l-16][K=4kk+2], [4kk+3].
// B layout (4x16 f32): VGPR0 = rows 4kk+koff, VGPR1 = rows 4kk+koff+1,
//                      n = lane & 15 in both halves.
// C/D layout: VGPR v -> row (v + 8*(lane>>4)), col = lane & 15.
// ---------------------------------------------------------------------------
__global__ void __launch_bounds__(256)
gcn_gemm_bias_relu(float*       __restrict__ hh_out,  // in: hidden_hat, out: result
                   const float* __restrict__ W,       // [96][96] row-major
                   const float* __restrict__ bias,    // [96]
                   int nRowBlocks) {
    int wid  = blockIdx.x * (blockDim.x >> 5) + (threadIdx.x >> 5);
    int lane = threadIdx.x & 31;
    if (wid >= nRowBlocks) return;

    const int m    = lane & 15;
    const int koff = (lane >> 4) * 2;   // 0 for lanes 0-15, 2 for lanes 16-31
    const int rb   = wid * 16;

    // Preload full A stripe: 24 chunks x (2 f32 per lane) = 16x96 f32 per wave.
    v2f a[24];
    const float* arow = hh_out + (size_t)(rb + m) * D_FEAT + koff;
#pragma unroll
    for (int kk = 0; kk < 24; ++kk) {
        a[kk] = *(const v2f*)(arow + kk * 4);   // 8B aligned: (row*96 + koff + 4kk) even
    }

#pragma unroll
    for (int ct = 0; ct < 6; ++ct) {            // six 16-col output tiles
        v8f acc = {};
        const float* Wb = W + ct * 16 + m;      // column within tile = lane & 15
#pragma unroll
        for (int kk = 0; kk < 24; ++kk) {
            const int krow = kk * 4 + koff;
            v2f b;
            b.x = Wb[(size_t)krow * D_FEAT];
            b.y = Wb[(size_t)(krow + 1) * D_FEAT];
            // D = A(16x4) * B(4x16) + C  -- f32 in, f32 out (matches reference)
            acc = __builtin_amdgcn_wmma_f32_16x16x4_f32(
                /*neg_a=*/false, a[kk], /*neg_b=*/false, b,
                /*c_mod=*/(short)0, acc, /*reuse_a=*/false, /*reuse_b=*/false);
        }

        const float bv     = bias[ct * 16 + m];
        const int   rowAdd = (lane >> 4) * 8;
#pragma unroll
        for (int v = 0; v < 8; ++v) {
            float o = acc[v] + bv;
            o = o > 0.f ? o : 0.f;
            hh_out[(size_t)(rb + v + rowAdd) * D_FEAT + ct * 16 + m] = o;
        }
    }
}

// ---------------------------------------------------------------------------
// Launch: inputs in setup_inputs() order: x, edge_w, weight, bias, row, col.
// d_out (50000*96 f32) doubles as hidden_hat scratch -> no d_ws dependency.
// ---------------------------------------------------------------------------
extern "C" void kernel_launch(void* const* d_in, const int* in_sizes, int n_in,
                              void* d_out, int out_size, void* d_ws, size_t ws_size,
                              hipStream_t stream) {
    const float* x    = (const float*)d_in[0];
    const float* ew   = (const float*)d_in[1];
    const float* W    = (const float*)d_in[2];
    const float* bias = (const float*)d_in[3];
    const int*   row  = (const int*)d_in[4];
    const int*   col  = (const int*)d_in[5];

    float* out = (float*)d_out;

    const int E = in_sizes[1];                 // 800000 edges
    const int N = in_sizes[0] / D_FEAT;        // 50000 nodes (divisible by 16)

    // 1) zero accumulator (d_out) every call -- deterministic under replay
    {
        int n4 = out_size / 4;
        int blocks = (n4 + 255) / 256;
        if (blocks > 4096) blocks = 4096;      // grid-stride covers the rest
        gcn_zero_f4<<<blocks, 256, 0, stream>>>((float4*)out, n4);
    }

    // 2) wave-per-edge gather/scale/atomic-scatter
    {
        long long threads = (long long)E * 32;
        int blocks = (int)((threads + 255) / 256);
        gcn_edge_scatter<<<blocks, 256, 0, stream>>>(x, ew, row, col, out, E);
    }

    // 3) in-place WMMA f32 GEMM + bias + ReLU
    {
        int nRowBlocks = N / 16;               // 3125
        int wavesPerBlock = 8;                 // 256 threads = 8 wave32
        int blocks = (nRowBlocks + wavesPerBlock - 1) / wavesPerBlock;
        gcn_gemm_bias_relu<<<blocks, 256, 0, stream>>>(out, W, bias, nRowBlocks);
    }
}